// PinyinGRUEmbeddings_3401614098990
// MI455X (gfx1250) — compile-verified
//
#include <hip/hip_runtime.h>

typedef float v2f __attribute__((ext_vector_type(2)));
typedef float v8f __attribute__((ext_vector_type(8)));

#define HDIM 8
#define TSTEPS 24
#define GDIM 24   // 3*H gate rows

// ---------------------------------------------------------------------------
// D[16x16] = A[16x4] * B[4x16] + C, full fp32 (V_WMMA_F32_16X16X4_F32)
// ---------------------------------------------------------------------------
__device__ __forceinline__ v8f wmma4(v2f a, v2f b, v8f c) {
    return __builtin_amdgcn_wmma_f32_16x16x4_f32(
        /*neg_a=*/false, a, /*neg_b=*/false, b,
        /*c_mod=*/(short)0, c, /*reuse_a=*/false, /*reuse_b=*/false);
}

// sigmoid via v_exp_f32 + v_rcp_f32 (no IEEE divide expansion)
__device__ __forceinline__ float sigm(float x) {
    return __builtin_amdgcn_rcpf(1.0f + __expf(-x));
}
__device__ __forceinline__ float tanh_f(float x) {
    return 2.0f * sigm(2.0f * x) - 1.0f;   // tanh(x) = 2*sigmoid(2x)-1
}

// A-layout tiles of a [24x8] weight matrix: [mtile][kchunk], each 16x4 fp32.
// A layout (ISA 7.12.2): lanes 0-15 = M rows; VGPR0 = K {0 | 2}, VGPR1 =
// K {1 | 3} for lane halves. Padding rows 24..31 of M-tile1 are filled with a
// DUPLICATE of the n-gate rows 16..23, so the D-tile's upper-half lanes
// compute valid n-gate pre-activations for their mirrored sample.
struct WT { v2f a[2][2]; };

__device__ __forceinline__ void load_wt(const float* __restrict__ W, int lane, WT& w) {
    const int m    = lane & 15;
    const int kofs = (lane < 16) ? 0 : 2;
#pragma unroll
    for (int mt = 0; mt < 2; ++mt) {
        int row = mt * 16 + m;
        if (row >= GDIM) row -= 8;            // rows 24..31 -> duplicate n rows 16..23
#pragma unroll
        for (int kc = 0; kc < 2; ++kc) {
            const int c0 = kc * 4 + kofs;
            v2f a;
            a.x = W[row * HDIM + c0];
            a.y = W[row * HDIM + c0 + 1];
            w.a[mt][kc] = a;
        }
    }
}

// C-layout bias seed with the same n-row duplication for M=24..31.
__device__ __forceinline__ v8f bias_tile(const float* __restrict__ b, int mt, int lane) {
    v8f c;
    const int base = mt * 16 + ((lane < 16) ? 0 : 8);
#pragma unroll
    for (int p = 0; p < 8; ++p) {
        int m = base + p;
        if (m >= GDIM) m -= 8;                // 24..31 -> n-gate bias 16..23
        c[p] = b[m];
    }
    return c;
}

// Build the two K-chunk B operands (K=0..3 / K=4..7) from a per-lane vec[8]
// valid on ALL lanes (lane and lane^16 mirror the same sample). Pure selects.
__device__ __forceinline__ void bchunks(const float v[HDIM], bool up, v2f& b0, v2f& b1) {
    b0.x = up ? v[2] : v[0];
    b0.y = up ? v[3] : v[1];
    b1.x = up ? v[6] : v[4];
    b1.y = up ? v[7] : v[5];
}

// Fused gate pre-activations for one layer:
//   g0 (r|z tile): W_ih*x + W_hh*h + (b_ih+b_hh), single 4-WMMA chain
//   g1i = W_ih*x + b_ih (n-gate input side), g1h = W_hh*h + b_hh (n-gate hidden side)
__device__ __forceinline__ void layer_preact(
    const WT& Wi, const WT& Wh, v2f bx0, v2f bx1, v2f bh0, v2f bh1,
    const v8f& seed0, const v8f& seed1i, const v8f& seed1h,
    v8f& g0, v8f& g1i, v8f& g1h)
{
    g0  = wmma4(Wi.a[0][0], bx0, seed0);
    g0  = wmma4(Wi.a[0][1], bx1, g0);
    g0  = wmma4(Wh.a[0][0], bh0, g0);
    g0  = wmma4(Wh.a[0][1], bh1, g0);
    g1i = wmma4(Wi.a[1][0], bx0, seed1i);
    g1i = wmma4(Wi.a[1][1], bx1, g1i);
    g1h = wmma4(Wh.a[1][0], bh0, seed1h);
    g1h = wmma4(Wh.a[1][1], bh1, g1h);
}

// GRU cell: g0 lanes0-15 = r preact, lanes16-31 = z preact. One half-swap
// shuffle gives every lane both r and z; both halves update h (all lanes valid).
__device__ __forceinline__ void gru_cell(float h[HDIM], const v8f& g0,
                                         const v8f& g1i, const v8f& g1h,
                                         int lane, bool up) {
    v8f rz, sw;
#pragma unroll
    for (int p = 0; p < 8; ++p) rz[p] = sigm(g0[p]);
#pragma unroll
    for (int p = 0; p < 8; ++p) sw[p] = __shfl(rz[p], lane ^ 16, 32);
#pragma unroll
    for (int p = 0; p < 8; ++p) {
        const float r = up ? sw[p] : rz[p];
        const float z = up ? rz[p] : sw[p];
        const float n = tanh_f(g1i[p] + r * g1h[p]);
        h[p] = n + z * (h[p] - n);            // (1-z)*n + z*h
    }
}

__global__ __launch_bounds__(128, 1) void pinyin_gru_wmma(
    const int* __restrict__ inputs, const float* __restrict__ emb,
    const float* __restrict__ w_ih1, const float* __restrict__ w_hh1,
    const float* __restrict__ b_ih1, const float* __restrict__ b_hh1,
    const float* __restrict__ w_ih2, const float* __restrict__ w_hh2,
    const float* __restrict__ b_ih2, const float* __restrict__ b_hh2,
    float* __restrict__ out, int Bn)
{
    const int  lane = threadIdx.x & 31;
    const bool up   = (lane >= 16);
    const int  wave = blockIdx.x * (blockDim.x >> 5) + (threadIdx.x >> 5);
    int sample = wave * 16 + (lane & 15);   // lanes 16..31 mirror samples 0..15
    if (sample >= Bn) sample = Bn - 1;      // clamp keeps EXEC full for WMMA

    WT W1i, W1h, W2i, W2h;
    load_wt(w_ih1, lane, W1i);
    load_wt(w_hh1, lane, W1h);
    load_wt(w_ih2, lane, W2i);
    load_wt(w_hh2, lane, W2h);

    // tile0 seeds are b_ih+b_hh (r/z fused); tile1 seeds stay separate (n gate)
    const v8f bi1_1 = bias_tile(b_ih1, 1, lane);
    const v8f bh1_1 = bias_tile(b_hh1, 1, lane);
    const v8f bi2_1 = bias_tile(b_ih2, 1, lane);
    const v8f bh2_1 = bias_tile(b_hh2, 1, lane);
    v8f b1_0 = bias_tile(b_ih1, 0, lane);
    v8f b2_0 = bias_tile(b_ih2, 0, lane);
    {
        const v8f t1 = bias_tile(b_hh1, 0, lane);
        const v8f t2 = bias_tile(b_hh2, 0, lane);
#pragma unroll
        for (int p = 0; p < 8; ++p) { b1_0[p] += t1[p]; b2_0[p] += t2[p]; }
    }

    float h1[HDIM], h2[HDIM];
#pragma unroll
    for (int p = 0; p < HDIM; ++p) { h1[p] = 0.0f; h2[p] = 0.0f; }

    const int* tok_ptr = inputs + (long)sample * TSTEPS;

    for (int t = 0; t < TSTEPS; ++t) {
        // ---- embedding gather (row 0 is the zero padding row in `emb`) ----
        const int tok = tok_ptr[t];
        const float4* er = (const float4*)(emb + tok * HDIM);
        const float4 e0 = er[0];
        const float4 e1 = er[1];
        v2f bx0, bx1;
        bx0.x = up ? e0.z : e0.x;  bx0.y = up ? e0.w : e0.y;
        bx1.x = up ? e1.z : e1.x;  bx1.y = up ? e1.w : e1.y;

        // ---- layer 1 ----
        v2f bh0, bh1;
        bchunks(h1, up, bh0, bh1);
        v8f g0, g1i, g1h;
        layer_preact(W1i, W1h, bx0, bx1, bh0, bh1, b1_0, bi1_1, bh1_1, g0, g1i, g1h);
        gru_cell(h1, g0, g1i, g1h, lane, up);

        // ---- layer 2 (x = h1; dropout inactive in eval) ----
        v2f c0, c1, ch0, ch1;
        bchunks(h1, up, c0, c1);
        bchunks(h2, up, ch0, ch1);
        layer_preact(W2i, W2h, c0, c1, ch0, ch1, b2_0, bi2_1, bh2_1, g0, g1i, g1h);
        gru_cell(h2, g0, g1i, g1h, lane, up);
    }

    if (lane < 16) {
        float4* o = (float4*)(out + (long)sample * HDIM);
        o[0] = make_float4(h2[0], h2[1], h2[2], h2[3]);
        o[1] = make_float4(h2[4], h2[5], h2[6], h2[7]);
    }
}

extern "C" void kernel_launch(void* const* d_in, const int* in_sizes, int n_in,
                              void* d_out, int out_size, void* d_ws, size_t ws_size,
                              hipStream_t stream) {
    const int*   inputs = (const int*)  d_in[0];
    const float* emb    = (const float*)d_in[1];
    const float* w_ih1  = (const float*)d_in[2];
    const float* w_hh1  = (const float*)d_in[3];
    const float* b_ih1  = (const float*)d_in[4];
    const float* b_hh1  = (const float*)d_in[5];
    const float* w_ih2  = (const float*)d_in[6];
    const float* w_hh2  = (const float*)d_in[7];
    const float* b_ih2  = (const float*)d_in[8];
    const float* b_hh2  = (const float*)d_in[9];
    float* out = (float*)d_out;

    const int Bn = in_sizes[0] / TSTEPS;           // 131072
    const int samplesPerBlock = 4 * 16;            // 128 threads = 4 waves x 16 samples
    const int grid = (Bn + samplesPerBlock - 1) / samplesPerBlock;

    pinyin_gru_wmma<<<grid, 128, 0, stream>>>(inputs, emb,
                                              w_ih1, w_hh1, b_ih1, b_hh1,
                                              w_ih2, w_hh2, b_ih2, b_hh2,
                                              out, Bn);
}